// Verifier_70489003262033
// MI455X (gfx1250) — compile-verified
//
#include <hip/hip_runtime.h>

#define D_DIM 1536
#define L_LAYERS 4
#define KTILES (D_DIM / 16)

typedef __attribute__((ext_vector_type(2))) float v2f;
typedef __attribute__((ext_vector_type(8))) float v8f;

__device__ __forceinline__ v8f wmma_f32(v2f a, v2f b, v8f c) {
  // (neg_a, A, neg_b, B, c_mod, C, reuse_a, reuse_b)
  return __builtin_amdgcn_wmma_f32_16x16x4_f32(false, a, false, b, (short)0, c,
                                               false, false);
}

// Async global->LDS 128b copy (GV mode, SADDR=off). Tracked by ASYNCcnt.
__device__ __forceinline__ void async_copy_b128(unsigned lds_off,
                                                const float* gaddr) {
  asm volatile("global_load_async_to_lds_b128 %0, %1, off"
               :: "v"(lds_off), "v"(gaddr)
               : "memory");
}
__device__ __forceinline__ void wait_async0() {
  asm volatile("s_wait_asynccnt 0x0" ::: "memory");
}

// ---------------------------------------------------------------------------
// lb0[n] = sum_d lb[d]*pos(lcs0[n,d]) + ub[d]*neg(lcs0[n,d]) + lc_bias0[n]
// ub0[n] = sum_d ub[d]*pos(ucs0[n,d]) + lb[d]*neg(ucs0[n,d]) + uc_bias0[n]
// One wave per output row; lane-strided coalesced reads.
// ---------------------------------------------------------------------------
__global__ __launch_bounds__(256) void init_bounds(
    const float* __restrict__ lb, const float* __restrict__ ub,
    const float* __restrict__ lcs0, const float* __restrict__ ucs0,
    const float* __restrict__ lcb0, const float* __restrict__ ucb0,
    float* __restrict__ lb0, float* __restrict__ ub0) {
  const int wave = threadIdx.x >> 5;
  const int lane = threadIdx.x & 31;
  const int n = blockIdx.x * 8 + wave;
  const float* lrow = lcs0 + (size_t)n * D_DIM;
  const float* urow = ucs0 + (size_t)n * D_DIM;
  float al = 0.f, au = 0.f;
  for (int d = lane; d < D_DIM; d += 32) {
    const float lbv = lb[d], ubv = ub[d];
    const float lc = lrow[d], uc = urow[d];
    al += lbv * fmaxf(lc, 0.f) + ubv * fminf(lc, 0.f);
    au += ubv * fmaxf(uc, 0.f) + lbv * fminf(uc, 0.f);
  }
#pragma unroll
  for (int off = 16; off; off >>= 1) {
    al += __shfl_xor(al, off, 32);
    au += __shfl_xor(au, off, 32);
  }
  if (lane == 0) {
    lb0[n] = al + lcb0[n];
    ub0[n] = au + ucb0[n];
  }
}

// ---------------------------------------------------------------------------
// Out = In^T (D x D), LDS-tiled 32x32.
// ---------------------------------------------------------------------------
__global__ __launch_bounds__(256) void transposeK(const float* __restrict__ In,
                                                  float* __restrict__ Out) {
  __shared__ float t[32][33];
  const int tx = threadIdx.x & 31;
  const int ty = threadIdx.x >> 5;
  const int x = blockIdx.x * 32 + tx;
  const int y = blockIdx.y * 32;
#pragma unroll
  for (int dy = 0; dy < 32; dy += 8)
    t[ty + dy][tx] = In[(size_t)(y + ty + dy) * D_DIM + x];
  __syncthreads();
  const int xo = blockIdx.y * 32 + tx;
  const int yo = blockIdx.x * 32;
#pragma unroll
  for (int dy = 0; dy < 32; dy += 8)
    Out[(size_t)(yo + ty + dy) * D_DIM + xo] = t[tx][ty + dy];
}

// ---------------------------------------------------------------------------
// Cout[a,b] = sum_c P[c,a]*pos(Cin[c,b]) + Q[c,a]*neg(Cin[c,b])
// Block tile 64(M) x 128(N), 128 threads = 4 waves in 2x2; wave tile 32x64 =
// 2x4 WMMA accumulators. K staged 16 deep in double-buffered LDS; P/Q staged
// by GLOBAL_LOAD_ASYNC_TO_LDS_B128 (ASYNCcnt), C staged through registers
// with the pos/neg split done once. One barrier per K-tile.
// ---------------------------------------------------------------------------
__global__ __launch_bounds__(128) void backsub_gemm(
    const float* __restrict__ P, const float* __restrict__ Q,
    const float* __restrict__ Cin, float* __restrict__ Cout) {
  __shared__ float sP[2][16][64];
  __shared__ float sQ[2][16][64];
  __shared__ float sCp[2][16][128];
  __shared__ float sCn[2][16][128];

  const int tid = threadIdx.x;
  const int lane = tid & 31;
  const int wave = tid >> 5;  // 0..3
  const int wm = wave >> 1;   // M half (a)
  const int wn = wave & 1;    // N half (b)
  const int aBase = blockIdx.y * 64;
  const int bBase = blockIdx.x * 128;
  const int hi = lane >> 4;   // 0 -> K{0,1}, 1 -> K{2,3}
  const int m15 = lane & 15;

  // staging slot assignment
  const int pcc = tid >> 4;          // 0..7  (row, +8 for second half)
  const int pgg = (tid & 15) << 2;   // 0..60 (float4 col group, 64-wide tile)
  const int cc0 = tid >> 5;          // 0..3  (row, +4 per quarter)
  const int cgg = (tid & 31) << 2;   // 0..124 (float4 col group, 128-wide)

  float4 rC[4];

  auto issue_async_PQ = [&](int ct, int buf) {
    const int crow = ct * 16;
#pragma unroll
    for (int h = 0; h < 2; ++h) {
      const int c = pcc + h * 8;
      const size_t go = (size_t)(crow + c) * D_DIM + aBase + pgg;
      async_copy_b128((unsigned)(uintptr_t)&sP[buf][c][pgg], P + go);
      async_copy_b128((unsigned)(uintptr_t)&sQ[buf][c][pgg], Q + go);
    }
  };
  auto load_C = [&](int ct) {
    const int crow = ct * 16;
#pragma unroll
    for (int q = 0; q < 4; ++q) {
      const int c = cc0 + q * 4;
      rC[q] = *(const float4*)(Cin + (size_t)(crow + c) * D_DIM + bBase + cgg);
    }
  };
  auto store_C = [&](int buf) {
#pragma unroll
    for (int q = 0; q < 4; ++q) {
      const int c = cc0 + q * 4;
      float4 cp, cn;
      cp.x = fmaxf(rC[q].x, 0.f); cn.x = fminf(rC[q].x, 0.f);
      cp.y = fmaxf(rC[q].y, 0.f); cn.y = fminf(rC[q].y, 0.f);
      cp.z = fmaxf(rC[q].z, 0.f); cn.z = fminf(rC[q].z, 0.f);
      cp.w = fmaxf(rC[q].w, 0.f); cn.w = fminf(rC[q].w, 0.f);
      *(float4*)&sCp[buf][c][cgg] = cp;
      *(float4*)&sCn[buf][c][cgg] = cn;
    }
  };

  v8f acc[2][4] = {};

  // prologue: stage tile 0 into buffer 0
  issue_async_PQ(0, 0);
  load_C(0);
  store_C(0);
  wait_async0();
  __syncthreads();

  for (int ct = 0; ct < KTILES; ++ct) {
    const int cur = ct & 1;
    const bool more = (ct + 1) < KTILES;
    if (more) {
      issue_async_PQ(ct + 1, cur ^ 1);  // async straight into LDS buf-next
      load_C(ct + 1);                   // latency hidden under the WMMAs below
      if (ct + 2 < KTILES) {
        const size_t off = (size_t)((ct + 2) * 16 + pcc) * D_DIM;
        __builtin_prefetch(Cin + off + bBase + cgg, 0, 1);
      }
    }

    // 4 WMMA K-steps of 4 over the 16-deep tile
#pragma unroll
    for (int kk = 0; kk < 4; ++kk) {
      const int c0 = kk * 4 + hi * 2;
      v2f aP[2], aQ[2], bP[4], bN[4];
#pragma unroll
      for (int i = 0; i < 2; ++i) {
        const int ar = wm * 32 + i * 16 + m15;
        aP[i][0] = sP[cur][c0][ar];     aP[i][1] = sP[cur][c0 + 1][ar];
        aQ[i][0] = sQ[cur][c0][ar];     aQ[i][1] = sQ[cur][c0 + 1][ar];
      }
#pragma unroll
      for (int j = 0; j < 4; ++j) {
        const int bc = wn * 64 + j * 16 + m15;
        bP[j][0] = sCp[cur][c0][bc];    bP[j][1] = sCp[cur][c0 + 1][bc];
        bN[j][0] = sCn[cur][c0][bc];    bN[j][1] = sCn[cur][c0 + 1][bc];
      }
#pragma unroll
      for (int i = 0; i < 2; ++i)
#pragma unroll
        for (int j = 0; j < 4; ++j) {
          acc[i][j] = wmma_f32(aP[i], bP[j], acc[i][j]);
          acc[i][j] = wmma_f32(aQ[i], bN[j], acc[i][j]);
        }
    }

    if (more) store_C(cur ^ 1);  // disjoint buffer: no barrier needed first
    wait_async0();               // buf-next fully resident before the barrier
    __syncthreads();
  }

  // epilogue: D layout — VGPR r: lanes<16 M=r, lanes>=16 M=r+8; N=lane&15
#pragma unroll
  for (int i = 0; i < 2; ++i)
#pragma unroll
    for (int j = 0; j < 4; ++j) {
      const int a0 = aBase + wm * 32 + i * 16 + hi * 8;
      const int b0 = bBase + wn * 64 + j * 16 + m15;
#pragma unroll
      for (int r = 0; r < 8; ++r)
        Cout[(size_t)(a0 + r) * D_DIM + b0] = acc[i][j][r];
    }
}

// ---------------------------------------------------------------------------
// newb[b] = sum_c bP[c]*pos(C[c,b]) + bQ[c]*neg(C[c,b]) + oldb[b]
// ---------------------------------------------------------------------------
__global__ __launch_bounds__(256) void bias_update(
    const float* __restrict__ bP, const float* __restrict__ bQ,
    const float* __restrict__ C, const float* __restrict__ oldb,
    float* __restrict__ newb) {
  const int b = blockIdx.x * 256 + threadIdx.x;
  float acc = 0.f;
  for (int c = 0; c < D_DIM; ++c) {
    const float v = C[(size_t)c * D_DIM + b];
    acc += bP[c] * fmaxf(v, 0.f) + bQ[c] * fminf(v, 0.f);
  }
  newb[b] = acc + oldb[b];
}

// ---------------------------------------------------------------------------
// out[n]     = sum_c lb0[c]*pos(Clc[c,n]) + ub0[c]*neg(Clc[c,n]) + lcb[n]
// out[D + n] = sum_c ub0[c]*pos(Cuc[c,n]) + lb0[c]*neg(Cuc[c,n]) + ucb[n]
// ---------------------------------------------------------------------------
__global__ __launch_bounds__(256) void final_bounds(
    const float* __restrict__ lb0, const float* __restrict__ ub0,
    const float* __restrict__ Clc, const float* __restrict__ Cuc,
    const float* __restrict__ lcb, const float* __restrict__ ucb,
    float* __restrict__ out) {
  const int n = blockIdx.x * 256 + threadIdx.x;
  float al = 0.f, au = 0.f;
  for (int c = 0; c < D_DIM; ++c) {
    const float l0 = lb0[c], u0 = ub0[c];
    const float vl = Clc[(size_t)c * D_DIM + n];
    const float vu = Cuc[(size_t)c * D_DIM + n];
    al += l0 * fmaxf(vl, 0.f) + u0 * fminf(vl, 0.f);
    au += u0 * fmaxf(vu, 0.f) + l0 * fminf(vu, 0.f);
  }
  out[n] = al + lcb[n];
  out[D_DIM + n] = au + ucb[n];
}

// ---------------------------------------------------------------------------
extern "C" void kernel_launch(void* const* d_in, const int* in_sizes, int n_in,
                              void* d_out, int out_size, void* d_ws,
                              size_t ws_size, hipStream_t stream) {
  const float* lb  = (const float*)d_in[0];
  const float* ub  = (const float*)d_in[1];
  const float* lcs = (const float*)d_in[2];
  const float* ucs = (const float*)d_in[3];
  const float* lcb = (const float*)d_in[4];
  const float* ucb = (const float*)d_in[5];
  float* out = (float*)d_out;

  const size_t D = D_DIM, DD = (size_t)D_DIM * D_DIM;
  float* ws = (float*)d_ws;
  float* lb0 = ws;              // D
  float* ub0 = ws + D;          // D
  float* bias = ws + 2 * D;     // 4 x D ping-pong bias buffers
  float* ucb_a = bias, * ucb_b = bias + D;
  float* lcb_a = bias + 2 * D, * lcb_b = bias + 3 * D;
  float* Cbase = ws + 6 * D;    // 4 x D*D ping-pong constraint matrices
  float* Cuc_a = Cbase, * Cuc_b = Cbase + DD;
  float* Clc_a = Cbase + 2 * DD, * Clc_b = Cbase + 3 * DD;

  // layer-0 forward bounds (all backsubs terminate on these)
  init_bounds<<<dim3(D_DIM / 8), dim3(256), 0, stream>>>(lb, ub, lcs, ucs, lcb,
                                                         ucb, lb0, ub0);

  // Only k = L-1 matters: earlier cur_lb/cur_ub iterations are dead code.
  const int k = L_LAYERS - 1;
  transposeK<<<dim3(48, 48), dim3(256), 0, stream>>>(lcs + k * DD, Clc_a);
  transposeK<<<dim3(48, 48), dim3(256), 0, stream>>>(ucs + k * DD, Cuc_a);
  hipMemcpyAsync(lcb_a, lcb + k * D, D * sizeof(float),
                 hipMemcpyDeviceToDevice, stream);
  hipMemcpyAsync(ucb_a, ucb + k * D, D * sizeof(float),
                 hipMemcpyDeviceToDevice, stream);

  float *CucCur = Cuc_a, *CucNxt = Cuc_b, *ClcCur = Clc_a, *ClcNxt = Clc_b;
  float *ucbCur = ucb_a, *ucbNxt = ucb_b, *lcbCur = lcb_a, *lcbNxt = lcb_b;

  const dim3 ggrid(D_DIM / 128, D_DIM / 64);
  for (int j = k - 1; j >= 0; --j) {
    // bias updates read the OLD C (same as the GEMMs), write new bias
    bias_update<<<dim3(D_DIM / 256), dim3(256), 0, stream>>>(
        ucb + j * D, lcb + j * D, CucCur, ucbCur, ucbNxt);
    bias_update<<<dim3(D_DIM / 256), dim3(256), 0, stream>>>(
        lcb + j * D, ucb + j * D, ClcCur, lcbCur, lcbNxt);
    // new_uc = ucs[j]^T @ pos(C_uc) + lcs[j]^T @ neg(C_uc)
    backsub_gemm<<<ggrid, dim3(128), 0, stream>>>(ucs + j * DD, lcs + j * DD,
                                                  CucCur, CucNxt);
    // new_lc = lcs[j]^T @ pos(C_lc) + ucs[j]^T @ neg(C_lc)
    backsub_gemm<<<ggrid, dim3(128), 0, stream>>>(lcs + j * DD, ucs + j * DD,
                                                  ClcCur, ClcNxt);
    float* t;
    t = CucCur; CucCur = CucNxt; CucNxt = t;
    t = ClcCur; ClcCur = ClcNxt; ClcNxt = t;
    t = ucbCur; ucbCur = ucbNxt; ucbNxt = t;
    t = lcbCur; lcbCur = lcbNxt; lcbNxt = t;
  }

  final_bounds<<<dim3(D_DIM / 256), dim3(256), 0, stream>>>(
      lb0, ub0, ClcCur, CucCur, lcbCur, ucbCur, out);

  (void)in_sizes; (void)n_in; (void)out_size; (void)ws_size;
}